// UST_54485955117639
// MI455X (gfx1250) — compile-verified
//
#include <hip/hip_runtime.h>
#include <hip/hip_bf16.h>
#include <math.h>

// ---- shapes from the reference ----
#define NB    8          // batch
#define NL    16         // frames
#define NPTS  128        // points per frame
#define NPB   2048       // points per batch (NL*NPTS)
#define P_TOTAL 16384    // NB*NPB
#define DIMF  256
#define TOKN  64
#define D1    257
#define BRN   3
#define NEG_INF -3.402823466e38f

// fast-GEMM tiling
#define FKC   32         // K chunk staged in LDS
#define APAD  36         // padded LDS row stride for A (conflict-free b64 reads)
#define BPAD  72         // padded LDS row stride for B (2*BPAD % 64 == 16)

#define USE_ASYNC_LDS 1  // GLOBAL_LOAD_ASYNC_TO_LDS_B128 path (fallback: sync float4 store)

typedef __attribute__((ext_vector_type(2))) float v2f;
typedef __attribute__((ext_vector_type(8))) float v8f;

__device__ __forceinline__ float act_apply(float x, int act) {
  if (act == 1) return x > 0.f ? x : 0.f;                       // relu
  if (act == 2) {                                               // tanh-approx gelu
    float x3 = x * x * x;
    float t = tanhf(0.7978845608028654f * (x + 0.044715f * x3));
    return 0.5f * x * (1.f + t);
  }
  if (act == 3) return x / (1.f + expf(-x));                    // silu
  return x;
}

// Async copy of 16B global -> LDS (per-lane), tracked by ASYNCcnt.
__device__ __forceinline__ void async_copy_b128(float* lds_dst, const float* gsrc) {
#if USE_ASYNC_LDS
  unsigned l = (unsigned)(uintptr_t)(__attribute__((address_space(3))) float*)lds_dst;
  asm volatile("global_load_async_to_lds_b128 %0, %1, off"
               :: "v"(l), "v"(gsrc) : "memory");
#else
  *(float4*)lds_dst = *(const float4*)gsrc;
#endif
}
__device__ __forceinline__ void async_wait0() {
#if USE_ASYNC_LDS
  asm volatile("s_wait_asynccnt 0x0" ::: "memory");
#endif
}

// ---------------------------------------------------------------------------
// FAST GEMM: requires M%128==0, N%64==0, K%32==0.
// 256 threads = 8 waves; block tile 128(M) x 64(N); wave tile 16 x 64.
// B chunks: double-buffered LDS filled by async global->LDS DMA (ASYNCcnt).
// A chunks: register-double-buffered b128 loads committed to LDS between
// barriers. Next chunk's DMA + loads are issued before the 32-WMMA body so
// memory latency hides behind v_wmma_f32_16x16x4_f32.
// ---------------------------------------------------------------------------
__global__ __launch_bounds__(256) void wmma_gemm_fast(
    const float* __restrict__ A, const float* __restrict__ W,
    const float* __restrict__ bias, float* __restrict__ C,
    int M, int N, int K, long sA, long sW, long sC, int act)
{
  __shared__ float As[128 * APAD];
  __shared__ float Bs[2][FKC * BPAD];
  A += (long)blockIdx.z * sA;
  W += (long)blockIdx.z * sW;
  C += (long)blockIdx.z * sC;
  const int mt   = blockIdx.x << 7;
  const int nt   = blockIdx.y << 6;
  const int tid  = threadIdx.x;
  const int lane = tid & 31;
  const int wm   = (tid >> 5) << 4;   // wave's 16-row offset inside block tile
  const int half = lane >> 4;
  const int l15  = lane & 15;

  // staging coordinates (uniform control flow, no guards needed)
  const int ar = tid >> 1;            // A row 0..127
  const int ac = (tid & 1) << 4;      // A col 0 or 16 (16 floats = 4x float4)
  const int br = tid >> 3;            // B row 0..31
  const int bc = (tid & 7) << 3;      // B col 0..56 (8 floats = 2x float4)

  const int nch = K / FKC;
  const float* gA = A + (long)(mt + ar) * K + ac;
  const float* gB = W + (long)br * N + nt + bc;
  const long gBstep = (long)FKC * N;

  v8f acc[4] = {};
  float4 r0, r1, r2, r3;

  // ---- prologue: chunk 0 ----
  async_copy_b128(&Bs[0][br * BPAD + bc],     gB);
  async_copy_b128(&Bs[0][br * BPAD + bc + 4], gB + 4);
  r0 = ((const float4*)gA)[0]; r1 = ((const float4*)gA)[1];
  r2 = ((const float4*)gA)[2]; r3 = ((const float4*)gA)[3];
  {
    float4* dA = (float4*)(As + ar * APAD + ac);
    dA[0] = r0; dA[1] = r1; dA[2] = r2; dA[3] = r3;
  }
  async_wait0();
  __syncthreads();

  int buf = 0;
  for (int i = 0; i < nch; ++i) {
    // issue next chunk's DMA + A loads before the compute body
    if (i + 1 < nch) {
      const float* gBn = gB + (long)(i + 1) * gBstep;
      async_copy_b128(&Bs[buf ^ 1][br * BPAD + bc],     gBn);
      async_copy_b128(&Bs[buf ^ 1][br * BPAD + bc + 4], gBn + 4);
      const float* gAn = gA + (i + 1) * FKC;
      r0 = ((const float4*)gAn)[0]; r1 = ((const float4*)gAn)[1];
      r2 = ((const float4*)gAn)[2]; r3 = ((const float4*)gAn)[3];
    }
    // ---- compute chunk i ----
    const float* Arow = As + (wm + l15) * APAD;
    const float* Bbuf = Bs[buf];
#pragma unroll
    for (int kk = 0; kk < FKC; kk += 4) {
      const int ak = kk + (half << 1);
      v2f a = *(const v2f*)(Arow + ak);
#pragma unroll
      for (int t = 0; t < 4; ++t) {
        v2f b;
        b.x = Bbuf[ak * BPAD + (t << 4) + l15];
        b.y = Bbuf[(ak + 1) * BPAD + (t << 4) + l15];
        acc[t] = __builtin_amdgcn_wmma_f32_16x16x4_f32(
            false, a, false, b, (short)0, acc[t], false, false);
      }
    }
    if (i + 1 == nch) break;
    async_wait0();
    __syncthreads();   // all waves done with chunk i; B_{i+1} landed
    float4* dA = (float4*)(As + ar * APAD + ac);
    dA[0] = r0; dA[1] = r1; dA[2] = r2; dA[3] = r3;
    __syncthreads();   // A_{i+1} visible
    buf ^= 1;
  }

#pragma unroll
  for (int t = 0; t < 4; ++t) {
    const int col = nt + (t << 4) + l15;
    const float bb = bias ? bias[col] : 0.f;
#pragma unroll
    for (int v = 0; v < 8; ++v) {
      const int row = mt + wm + v + (half << 3);
      C[(long)row * N + col] = act_apply(acc[t][v] + bb, act);
    }
  }
}

// ---------------------------------------------------------------------------
// GENERAL GEMM (bounds-checked) for ragged shapes: M=8 FCs, K=4 pos layer,
// N=257 / N=60 layers. One wave per 16x64 strip.
// ---------------------------------------------------------------------------
__global__ __launch_bounds__(32) void wmma_gemm(
    const float* __restrict__ A, const float* __restrict__ W,
    const float* __restrict__ bias, float* __restrict__ C,
    int M, int N, int K, long sA, long sW, long sC, int act)
{
  A += (long)blockIdx.z * sA;
  W += (long)blockIdx.z * sW;
  C += (long)blockIdx.z * sC;
  const int mt   = blockIdx.x << 4;
  const int nt   = blockIdx.y << 6;
  const int lane = threadIdx.x;
  const int half = lane >> 4;
  const int l15  = lane & 15;
  const int am   = mt + l15;
  const bool arow = (am < M);

  v8f acc[4] = {};
  for (int k0 = 0; k0 < K; k0 += 4) {
    const int ak = k0 + (half << 1);
    v2f a;
    a.x = (arow && ak     < K) ? A[(long)am * K + ak    ] : 0.f;
    a.y = (arow && ak + 1 < K) ? A[(long)am * K + ak + 1] : 0.f;
#pragma unroll
    for (int t = 0; t < 4; ++t) {
      const int bn = nt + (t << 4) + l15;
      v2f b;
      b.x = (ak     < K && bn < N) ? W[(long)(ak    ) * N + bn] : 0.f;
      b.y = (ak + 1 < K && bn < N) ? W[(long)(ak + 1) * N + bn] : 0.f;
      acc[t] = __builtin_amdgcn_wmma_f32_16x16x4_f32(
          false, a, false, b, (short)0, acc[t], false, false);
    }
  }
#pragma unroll
  for (int t = 0; t < 4; ++t) {
    const int col = nt + (t << 4) + l15;
    if (col >= N) continue;
    const float bb = bias ? bias[col] : 0.f;
#pragma unroll
    for (int v = 0; v < 8; ++v) {
      const int row = mt + v + (half << 3);
      if (row < M) C[(long)row * N + col] = act_apply(acc[t][v] + bb, act);
    }
  }
}

// column-wise max over `rows` rows per batch (blockIdx.y = batch)
__global__ void maxpool_ch(const float* __restrict__ X, float* __restrict__ Y,
                           int rows, int cols) {
  int c = blockIdx.x * blockDim.x + threadIdx.x;
  int b = blockIdx.y;
  if (c >= cols) return;
  const float* Xb = X + (long)b * rows * cols;
  float m = NEG_INF;
  for (int r = 0; r < rows; ++r) m = fmaxf(m, Xb[(long)r * cols + c]);
  Y[(long)b * cols + c] = m;
}

// full_emb = embB[64,32] @ tok1[32,1]  -> [64]
__global__ void emb_kernel(const float* __restrict__ embB,
                           const float* __restrict__ tok1,
                           float* __restrict__ fe) {
  int t = threadIdx.x;
  if (t < TOKN) {
    float s = 0.f;
    for (int j = 0; j < 32; ++j) s += embB[t * 32 + j] * tok1[j];
    fe[t] = s;
  }
}

// idx[p] = argmax_t( log_softmax(logits)[t] + gumbel[p,t] )
__global__ void route_kernel(const float* __restrict__ logits,
                             const float* __restrict__ gumbel,
                             int* __restrict__ idx) {
  int p = blockIdx.x * blockDim.x + threadIdx.x;
  if (p >= P_TOTAL) return;
  const float* L = logits + (long)p * TOKN;
  const float* G = gumbel + (long)p * TOKN;
  float mx = NEG_INF;
  for (int t = 0; t < TOKN; ++t) mx = fmaxf(mx, L[t]);
  float s = 0.f;
  for (int t = 0; t < TOKN; ++t) s += expf(L[t] - mx);
  float lse = mx + logf(s);
  float best = NEG_INF; int bi = 0;
  for (int t = 0; t < TOKN; ++t) {
    float z = L[t] - lse + G[t];
    if (z > best) { best = z; bi = t; }
  }
  idx[p] = bi;
}

// featcat[p, 0:256] = features[p]; featcat[p, 256] = full_emb[idx[p]]
__global__ void concat_kernel(const float* __restrict__ feats,
                              const float* __restrict__ fe,
                              const int* __restrict__ idx,
                              float* __restrict__ out) {
  long i = (long)blockIdx.x * blockDim.x + threadIdx.x;
  if (i >= (long)P_TOTAL * D1) return;
  int p = (int)(i / D1), c = (int)(i % D1);
  out[i] = (c < DIMF) ? feats[(long)p * DIMF + c] : fe[idx[p]];
}

// stable counting sort of idx within each batch (keys in [0,64))
__global__ void sort_kernel(const int* __restrict__ idx, int* __restrict__ order) {
  int b = threadIdx.x;
  if (b >= NB) return;
  int hist[TOKN];
  for (int t = 0; t < TOKN; ++t) hist[t] = 0;
  const int* ib = idx + b * NPB;
  for (int j = 0; j < NPB; ++j) hist[ib[j]]++;
  int off[TOKN]; int run = 0;
  for (int t = 0; t < TOKN; ++t) { off[t] = run; run += hist[t]; }
  int* ob = order + b * NPB;
  for (int j = 0; j < NPB; ++j) ob[off[ib[j]]++] = j;
}

__global__ void gather_feat(const float* __restrict__ fc,
                            const int* __restrict__ order,
                            float* __restrict__ out) {
  long i = (long)blockIdx.x * blockDim.x + threadIdx.x;
  if (i >= (long)P_TOTAL * D1) return;
  int pd = (int)(i / D1), c = (int)(i % D1);
  int b = pd / NPB;
  int src = b * NPB + order[pd];
  out[i] = fc[(long)src * D1 + c];
}

// xyzt[p, 0:3] = xyz[order]; xyzt[p, 3] = floor((li+1)/NL)
__global__ void gather_xyzt(const float* __restrict__ xyzs,
                            const int* __restrict__ order,
                            float* __restrict__ xyzt) {
  int i = blockIdx.x * blockDim.x + threadIdx.x;
  if (i >= P_TOTAL * 4) return;
  int p = i >> 2, c = i & 3;
  int b = p / NPB, j = p % NPB;
  if (c < 3) {
    int src = b * NPB + order[p];
    xyzt[i] = xyzs[(long)src * 3 + c];
  } else {
    int li = j / NPTS;
    xyzt[i] = (float)((li + 1) / NL);
  }
}

// branch max with the torch reshape scramble on the feature operand.
// fsorted: [B, NPB, D1]; pos: [B, NL, NPTS, D1]; out: [B, BRN*D1]
__global__ void branch_max(const float* __restrict__ fsorted,
                           const float* __restrict__ pos,
                           float* __restrict__ out) {
  int c  = blockIdx.x * blockDim.x + threadIdx.x;
  int br = blockIdx.y;
  int b  = blockIdx.z;
  if (c >= D1) return;
  int w = NL >> br;          // 16, 8, 4
  int R = w * NPTS;
  const float* fb = fsorted + (long)b * NPB * D1;
  const float* pb = pos     + (long)b * NPB * D1;
  float m = NEG_INF;
  for (int r = 0; r < R; ++r) {
    int flat = r * D1 + c;
    int wi  = flat / (D1 * NPTS);
    int rem = flat % (D1 * NPTS);
    int d   = rem / NPTS;
    int ni  = rem % NPTS;
    int li  = wi << br;                      // idxs[wi]
    float fv = fb[(long)li * (D1 * NPTS) + d * NPTS + ni];
    int lip = (r / NPTS) << br;
    float pv = pb[((long)lip * NPTS + (r % NPTS)) * D1 + c];
    m = fmaxf(m, fv + pv);
  }
  out[(long)b * (BRN * D1) + br * D1 + c] = m;
}

__global__ void ln_kernel(const float* __restrict__ x, const float* __restrict__ g,
                          const float* __restrict__ beta, float* __restrict__ y) {
  int b = threadIdx.x;
  if (b >= NB) return;
  const int C = BRN * D1;   // 771
  const float* xb = x + b * C;
  float mu = 0.f;
  for (int c = 0; c < C; ++c) mu += xb[c];
  mu /= (float)C;
  float var = 0.f;
  for (int c = 0; c < C; ++c) { float d = xb[c] - mu; var += d * d; }
  var /= (float)C;
  float inv = rsqrtf(var + 1e-5f);
  float* yb = y + b * C;
  for (int c = 0; c < C; ++c) yb[c] = (xb[c] - mu) * inv * g[c] + beta[c];
}

extern "C" void kernel_launch(void* const* d_in, const int* in_sizes, int n_in,
                              void* d_out, int out_size, void* d_ws, size_t ws_size,
                              hipStream_t stream) {
  (void)in_sizes; (void)n_in; (void)out_size; (void)ws_size;
  // Input order: top-level insertion order (xyzs, features, gumbel, params),
  // params flattened as a pytree with dict keys sorted alphabetically.
  const float* xyzs   = (const float*)d_in[0];
  const float* feats  = (const float*)d_in[1];
  const float* gumbel = (const float*)d_in[2];
  const float* embB   = (const float*)d_in[3];
  const float* h_b    = (const float*)d_in[4];
  const float* h_b1   = (const float*)d_in[5];
  const float* h_b2   = (const float*)d_in[6];
  const float* h_g    = (const float*)d_in[7];
  const float* h_w1   = (const float*)d_in[8];   // [771,1024]
  const float* h_w2   = (const float*)d_in[9];   // [1024,60]
  const float* p_b1   = (const float*)d_in[10];
  const float* p_b2   = (const float*)d_in[11];
  const float* p_w1   = (const float*)d_in[12];  // [4,128]
  const float* p_w2   = (const float*)d_in[13];  // [128,257]
  const float* pn_c1b = (const float*)d_in[14];
  const float* pn_c1w = (const float*)d_in[15];  // [256,64]
  const float* pn_c2b = (const float*)d_in[16];
  const float* pn_c2w = (const float*)d_in[17];  // [64,128]
  const float* pn_c3b = (const float*)d_in[18];
  const float* pn_c3w = (const float*)d_in[19];  // [128,1024]
  const float* pn_f1b = (const float*)d_in[20];
  const float* pn_f1w = (const float*)d_in[21];  // [1024,512]
  const float* pn_f2b = (const float*)d_in[22];
  const float* pn_f2w = (const float*)d_in[23];  // [512,256]
  const float* pn_f3b = (const float*)d_in[24];
  const float* pn_f3w = (const float*)d_in[25];  // [256,64]
  const float* tn_c1b = (const float*)d_in[26];
  const float* tn_c1w = (const float*)d_in[27];
  const float* tn_c2b = (const float*)d_in[28];
  const float* tn_c2w = (const float*)d_in[29];
  const float* tn_c3b = (const float*)d_in[30];
  const float* tn_c3w = (const float*)d_in[31];
  const float* tn_f1b = (const float*)d_in[32];
  const float* tn_f1w = (const float*)d_in[33];
  const float* tn_f2b = (const float*)d_in[34];
  const float* tn_f2w = (const float*)d_in[35];
  const float* tn_f3b = (const float*)d_in[36];  // [65536]
  const float* tn_f3w = (const float*)d_in[37];  // [256,65536]
  const float* tok1   = (const float*)d_in[38];  // [32,1]

  // ---- workspace layout ----
  float* ws     = (float*)d_ws;
  float* bufA   = ws;                                  // P*1024
  float* bufB   = bufA + (long)P_TOTAL * 1024;         // P*1024
  float* trans  = bufB + (long)P_TOTAL * 1024;         // B*65536
  float* g0     = trans + (long)NB * 65536;            // B*1024
  float* g1     = g0 + NB * 1024;                      // B*512
  float* g2     = g1 + NB * 512;                       // B*256
  float* femb   = g2 + NB * 256;                       // 64
  float* xyzt   = femb + 64;                           // P*4
  float* out771 = xyzt + (long)P_TOTAL * 4;            // B*771
  float* lnout  = out771 + NB * 771;                   // B*771
  float* headh  = lnout + NB * 771;                    // B*1024
  int*   idxbuf = (int*)(headh + NB * 1024);           // P ints
  int*   ordbuf = idxbuf + P_TOTAL;                    // P ints

  auto gemm = [&](const float* A, const float* W, const float* bias, float* C,
                  int M, int N, int K, long sA, long sW, long sC, int nb, int act) {
    if ((M % 128) == 0 && (N % 64) == 0 && (K % FKC) == 0) {
      dim3 grid(M / 128, N / 64, nb);
      wmma_gemm_fast<<<grid, 256, 0, stream>>>(A, W, bias, C, M, N, K, sA, sW, sC, act);
    } else {
      dim3 grid((M + 15) / 16, (N + 63) / 64, nb);
      wmma_gemm<<<grid, 32, 0, stream>>>(A, W, bias, C, M, N, K, sA, sW, sC, act);
    }
  };

  // ---- tnet ----
  gemm(feats, tn_c1w, tn_c1b, bufA, P_TOTAL, 64,   DIMF, 0, 0, 0, 1, 1);
  gemm(bufA,  tn_c2w, tn_c2b, bufB, P_TOTAL, 128,  64,   0, 0, 0, 1, 1);
  gemm(bufB,  tn_c3w, tn_c3b, bufA, P_TOTAL, 1024, 128,  0, 0, 0, 1, 1);
  maxpool_ch<<<dim3(4, NB), 256, 0, stream>>>(bufA, g0, NPB, 1024);
  gemm(g0, tn_f1w, tn_f1b, g1, NB, 512, 1024, 0, 0, 0, 1, 1);
  gemm(g1, tn_f2w, tn_f2b, g2, NB, 256, 512,  0, 0, 0, 1, 1);
  gemm(g2, tn_f3w, tn_f3b, trans, NB, 65536, 256, 0, 0, 0, 1, 0);
  // x = einsum('bnc,bcd->bnd', feats, trans): batched GEMM
  gemm(feats, trans, nullptr, bufB, NPB, DIMF, DIMF,
       (long)NPB * DIMF, 65536, (long)NPB * DIMF, NB, 0);

  // ---- pointnet main stack ----
  gemm(bufB, pn_c1w, pn_c1b, bufA, P_TOTAL, 64,   DIMF, 0, 0, 0, 1, 1);
  gemm(bufA, pn_c2w, pn_c2b, bufB, P_TOTAL, 128,  64,   0, 0, 0, 1, 1);
  gemm(bufB, pn_c3w, pn_c3b, bufA, P_TOTAL, 1024, 128,  0, 0, 0, 1, 1);
  gemm(bufA, pn_f1w, pn_f1b, bufB, P_TOTAL, 512,  1024, 0, 0, 0, 1, 1);
  gemm(bufB, pn_f2w, pn_f2b, bufA, P_TOTAL, 256,  512,  0, 0, 0, 1, 1);
  gemm(bufA, pn_f3w, pn_f3b, bufB, P_TOTAL, TOKN, 256,  0, 0, 0, 1, 0); // logits

  // ---- routing / prompt / sort ----
  emb_kernel<<<1, 64, 0, stream>>>(embB, tok1, femb);
  route_kernel<<<(P_TOTAL + 255) / 256, 256, 0, stream>>>(bufB, gumbel, idxbuf);
  sort_kernel<<<1, NB, 0, stream>>>(idxbuf, ordbuf);
  {
    long tot = (long)P_TOTAL * D1;
    concat_kernel<<<(unsigned)((tot + 255) / 256), 256, 0, stream>>>(feats, femb, idxbuf, bufA);
    gather_feat<<<(unsigned)((tot + 255) / 256), 256, 0, stream>>>(bufA, ordbuf, bufB);
  }
  gather_xyzt<<<(P_TOTAL * 4 + 255) / 256, 256, 0, stream>>>(xyzs, ordbuf, xyzt);

  // ---- positional MLP (reuse disjoint regions of bufA) ----
  float* posh = bufA;                              // P*128
  float* pos  = bufA + (long)P_TOTAL * 512;        // P*257
  gemm(xyzt, p_w1, p_b1, posh, P_TOTAL, 128, 4,   0, 0, 0, 1, 3); // silu (K=4 -> general)
  gemm(posh, p_w2, p_b2, pos,  P_TOTAL, D1,  128, 0, 0, 0, 1, 0); // N=257 -> general

  // ---- branch max + head ----
  branch_max<<<dim3((D1 + 63) / 64, BRN, NB), 64, 0, stream>>>(bufB, pos, out771);
  ln_kernel<<<1, NB, 0, stream>>>(out771, h_g, h_b, lnout);
  gemm(lnout, h_w1, h_b1, headh, NB, 1024, BRN * D1, 0, 0, 0, 1, 2); // gelu
  gemm(headh, h_w2, h_b2, (float*)d_out, NB, 60, 1024, 0, 0, 0, 1, 0);
}